// ContrastiveTokenLoss_51702816309253
// MI455X (gfx1250) — compile-verified
//
#include <hip/hip_runtime.h>
#include <hip/hip_bf16.h>
#include <math.h>

// Problem constants (fixed by setup_inputs in the reference):
// B=2, T=2048, V=32000, CT_LENGTH=0.25 -> ct_len=512, PRECED_M_NEGATIVES=0.5 -> win=256
#define CTL_B      2
#define CTL_T      2048
#define CTL_V      32000
#define CTL_CT     512
#define CTL_WIN    256
#define CTL_IGNORE (-100)
#define CTL_PAD    0
#define CTL_ROWS   (CTL_B * CTL_CT)   // 1024

// ---------------- CDNA5 async global->LDS helpers (probe-guarded) ----------------
#if defined(__HIP_DEVICE_COMPILE__) && defined(__gfx1250__)
#define CTL_GFX1250 1
#endif

#if defined(__has_builtin)
#  if __has_builtin(__builtin_amdgcn_global_load_async_to_lds_b32)
#    define CTL_HAVE_ASYNC_BUILTIN 1
#  endif
#  if __has_builtin(__builtin_amdgcn_s_wait_asynccnt)
#    define CTL_HAVE_WAIT_BUILTIN 1
#  endif
#endif

__device__ __forceinline__ void ctl_async_ld_b32(const int* g, int* l) {
#if defined(CTL_HAVE_ASYNC_BUILTIN)
    // Params are typed: (addrspace(1) int* src, addrspace(3) int* dst, imm offset, imm cpol)
    __builtin_amdgcn_global_load_async_to_lds_b32(
        (__attribute__((address_space(1))) int*)g,
        (__attribute__((address_space(3))) int*)l, 0, 0);
#elif defined(CTL_GFX1250)
    // Inline-asm form per CDNA5 ISA: VDST = per-lane LDS byte address,
    // VADDR = 64-bit global address, GV mode (saddr = off).
    unsigned loff = (unsigned)(unsigned long long)
        (__attribute__((address_space(3))) int*)l;
    asm volatile("global_load_async_to_lds_b32 %0, %1, off"
                 :: "v"(loff), "v"(g) : "memory");
#else
    *l = *g;  // host-pass / non-gfx1250 fallback (never executed on device)
#endif
}

__device__ __forceinline__ void ctl_wait_async0() {
#if defined(CTL_HAVE_WAIT_BUILTIN)
    __builtin_amdgcn_s_wait_asynccnt(0);
#elif defined(CTL_GFX1250)
    asm volatile("s_wait_asynccnt 0x0" ::: "memory");
#endif
}

// ---------------- Kernel 1: one block per (b, i) row ----------------
// blockDim.x == CTL_WIN (256): thread t handles negative position j = i - 256 + t.
__global__ __launch_bounds__(CTL_WIN)
void ctl_row_kernel(const float* __restrict__ logits,   // (B, T, V) f32
                    const int*   __restrict__ target,   // (B, T) i32
                    float* __restrict__ row_loss,       // (CTL_ROWS)
                    float* __restrict__ row_valid)      // (CTL_ROWS)
{
    __shared__ int   s_tgt[CTL_WIN];
    __shared__ float s_part[CTL_WIN / 32];

    const int row = blockIdx.x;            // 0 .. CTL_ROWS-1
    const int b   = row / CTL_CT;
    const int i   = row - b * CTL_CT;
    const int t   = threadIdx.x;           // 0 .. 255

    const int* tgt_b = target + b * CTL_T;

    // Stage the target window tgt[b, i-256 : i] into LDS via the CDNA5
    // async global->LDS path (tracked with ASYNCcnt).
    const int j  = i - CTL_WIN + t;        // j < i always; may be negative
    const int jc = (j < 0) ? 0 : j;        // clamp for a safe (discarded) load
    ctl_async_ld_b32(tgt_b + jc, &s_tgt[t]);
    ctl_wait_async0();
    __syncthreads();

    const int  tj        = s_tgt[t];
    const int  ti        = tgt_b[i];                 // uniform per block
    const bool valid_tgt = (ti != CTL_IGNORE);

    int ti_safe = valid_tgt ? ti : CTL_PAD;
    if (ti_safe < 0) ti_safe = 0; else if (ti_safe >= CTL_V) ti_safe = CTL_V - 1;

    const float* lrow = logits + ((long long)b * CTL_T + i) * (long long)CTL_V;
    const float  pos  = lrow[ti_safe];               // broadcast gather

    float e = 0.0f;
    if ((j >= 0) && (tj != CTL_PAD) && valid_tgt) {
        int tjc = tj;                                // JAX gather clips OOB
        if (tjc < 0) tjc = 0; else if (tjc >= CTL_V) tjc = CTL_V - 1;
        e = __expf(lrow[tjc] - pos);                 // the one hot gather
    }

    // wave32 tree reduction, then combine 8 wave partials in LDS.
    #pragma unroll
    for (int off = 16; off > 0; off >>= 1) e += __shfl_down(e, off, 32);

    const int wave = t >> 5, lane = t & 31;
    if (lane == 0) s_part[wave] = e;
    __syncthreads();

    if (t == 0) {
        float s = 0.0f;
        #pragma unroll
        for (int w = 0; w < CTL_WIN / 32; ++w) s += s_part[w];
        row_loss[row]  = valid_tgt ? log1pf(s) : 0.0f;
        row_valid[row] = valid_tgt ? 1.0f : 0.0f;
    }
}

// ---------------- Kernel 2: deterministic final reduction ----------------
__global__ __launch_bounds__(CTL_ROWS)
void ctl_reduce_kernel(const float* __restrict__ row_loss,
                       const float* __restrict__ row_valid,
                       float* __restrict__ out)
{
    __shared__ float sl[CTL_ROWS / 32];
    __shared__ float sv[CTL_ROWS / 32];

    const int t = threadIdx.x;             // 0 .. 1023, exactly CTL_ROWS
    float l = row_loss[t];
    float v = row_valid[t];

    #pragma unroll
    for (int off = 16; off > 0; off >>= 1) {
        l += __shfl_down(l, off, 32);
        v += __shfl_down(v, off, 32);
    }
    const int wave = t >> 5, lane = t & 31;
    if (lane == 0) { sl[wave] = l; sv[wave] = v; }
    __syncthreads();

    if (wave == 0) {
        l = sl[lane];
        v = sv[lane];
        #pragma unroll
        for (int off = 16; off > 0; off >>= 1) {
            l += __shfl_down(l, off, 32);
            v += __shfl_down(v, off, 32);
        }
        if (lane == 0) out[0] = l / fmaxf(v, 1.0f);
    }
}

extern "C" void kernel_launch(void* const* d_in, const int* in_sizes, int n_in,
                              void* d_out, int out_size, void* d_ws, size_t ws_size,
                              hipStream_t stream) {
    (void)in_sizes; (void)n_in; (void)out_size; (void)ws_size;
    const float* logits = (const float*)d_in[0];   // (B, T, V) float32
    const int*   target = (const int*)d_in[1];     // (B, T) int32 (per harness dtype rule)

    float* row_loss  = (float*)d_ws;               // CTL_ROWS floats
    float* row_valid = row_loss + CTL_ROWS;        // CTL_ROWS floats (8 KB total)

    ctl_row_kernel<<<CTL_ROWS, CTL_WIN, 0, stream>>>(logits, target, row_loss, row_valid);
    ctl_reduce_kernel<<<1, CTL_ROWS, 0, stream>>>(row_loss, row_valid, (float*)d_out);
}